// QuantumClassifier_6725918785876
// MI455X (gfx1250) — compile-verified
//
#include <hip/hip_runtime.h>
#include <hip/hip_bf16.h>

typedef __attribute__((ext_vector_type(2))) float v2f;
typedef __attribute__((ext_vector_type(8))) float v8f;

struct C2 { float x, y; };
__device__ inline C2 cmul(C2 a, C2 b){ return {a.x*b.x - a.y*b.y, a.x*b.y + a.y*b.x}; }
__device__ inline C2 cadd(C2 a, C2 b){ return {a.x+b.x, a.y+b.y}; }

// ---------------------------------------------------------------------------
// Setup: build A = Re(M^dag Z M), where M = CNOT_ring * (tensor of Rot gates).
// One small block; thread j simulates basis column e_j through the constant
// part of the circuit, then A is formed from the columns.
// ---------------------------------------------------------------------------
__global__ void qc_setup_kernel(const float* __restrict__ w, float* __restrict__ Ag) {
  __shared__ C2 cols[16][16];   // cols[j][k] = (M e_j)[k]
  const int j = threadIdx.x;
  if (j < 16) {
    C2 st[16];
    #pragma unroll
    for (int k = 0; k < 16; k++) st[k] = {0.f, 0.f};
    st[j] = {1.f, 0.f};
    // Rot(phi,theta,omega) = RZ(omega) RY(theta) RZ(phi) on each wire
    for (int q = 0; q < 4; q++) {
      const float phi = w[q*3+0], th = w[q*3+1], om = w[q*3+2];
      float sth, cth; __sincosf(0.5f*th, &sth, &cth);
      const float ap = 0.5f*(phi+om), am = 0.5f*(phi-om);
      float sp, cp; __sincosf(ap, &sp, &cp);
      float sm, cm; __sincosf(am, &sm, &cm);
      const C2 u00 = {  cp*cth, -sp*cth };  //  e^{-i(phi+om)/2} cos
      const C2 u01 = { -cm*sth, -sm*sth };  // -e^{+i(phi-om)/2} sin
      const C2 u10 = {  cm*sth, -sm*sth };  //  e^{-i(phi-om)/2} sin
      const C2 u11 = {  cp*cth,  sp*cth };  //  e^{+i(phi+om)/2} cos
      const int bit = 8 >> q;               // wire 0 is the MSB of the flat index
      for (int base = 0; base < 16; base++) {
        if (base & bit) continue;
        const C2 a = st[base], b = st[base | bit];
        st[base]       = cadd(cmul(u00, a), cmul(u01, b));
        st[base | bit] = cadd(cmul(u10, a), cmul(u11, b));
      }
    }
    // CNOT ring, r = 1: (0,1),(1,2),(2,3),(3,0); out[idx] = in[idx ^ (ctrl? bt:0)]
    for (int i = 0; i < 4; i++) {
      const int bc = 8 >> i, bt = 8 >> ((i+1) & 3);
      C2 tmp[16];
      for (int idx = 0; idx < 16; idx++) {
        const int src = (idx & bc) ? (idx ^ bt) : idx;
        tmp[idx] = st[src];
      }
      for (int idx = 0; idx < 16; idx++) st[idx] = tmp[idx];
    }
    for (int k = 0; k < 16; k++) cols[j][k] = st[k];
  }
  __syncthreads();
  if (j < 16) {
    for (int i = 0; i < 16; i++) {
      float acc = 0.f;
      for (int k = 0; k < 16; k++) {
        const float z = (k < 8) ? 1.f : -1.f;
        acc += z * (cols[i][k].x * cols[j][k].x + cols[i][k].y * cols[j][k].y);
      }
      Ag[i*16 + j] = acc;   // A is symmetric; row-major A[k][n]
    }
  }
}

// ---------------------------------------------------------------------------
// Main: out[b] = s_b^T A s_b via WMMA f32 16x16x4.
// Each wave handles 32 batch rows = two 16x16 C-tiles, K = 16 (4 WMMA steps).
// ---------------------------------------------------------------------------
#define WPB 8   // waves per 256-thread block

__global__ void __launch_bounds__(256) qc_main_kernel(const float* __restrict__ x,
                                                      const float* __restrict__ Ag,
                                                      float* __restrict__ out) {
  __shared__ __align__(16) float Sbuf[WPB][32 * 16];
  __shared__ __align__(16) float Pbuf[WPB][32 * 16];
  const int lane = threadIdx.x & 31;
  const int wv   = threadIdx.x >> 5;
  const int b    = blockIdx.x * blockDim.x + threadIdx.x;

  // ---- per-lane product state s (real, 16 entries), kept in registers ----
  const float4 xv = reinterpret_cast<const float4*>(x)[b];
  float s0,c0,s1,c1,s2,c2,s3,c3;
  __sincosf(0.5f*xv.x, &s0, &c0);
  __sincosf(0.5f*xv.y, &s1, &c1);
  __sincosf(0.5f*xv.z, &s2, &c2);
  __sincosf(0.5f*xv.w, &s3, &c3);
  const float p01[4] = {c0*c1, c0*s1, s0*c1, s0*s1};
  const float p23[4] = {c2*c3, c2*s3, s2*c3, s2*s3};
  float sv[16];
  #pragma unroll
  for (int i = 0; i < 4; i++)
    #pragma unroll
    for (int k = 0; k < 4; k++)
      sv[i*4 + k] = p01[i] * p23[k];   // flat index = (q0*2+q1)*4 + (q2*2+q3)

  // ---- stage S tile in LDS: row = lane (32 rows per wave) ----
  float* Sw = &Sbuf[wv][0];
  #pragma unroll
  for (int i = 0; i < 4; i++)
    reinterpret_cast<float4*>(&Sw[lane*16])[i] =
        make_float4(sv[4*i], sv[4*i+1], sv[4*i+2], sv[4*i+3]);
  asm volatile("" ::: "memory");   // keep compiler from hoisting LDS reads above stores

  // ---- GEMM: P(32x16) = S(32x16) * A(16x16), two 16x16 tiles, K split by 4 ----
  v8f acc0 = {0,0,0,0,0,0,0,0};
  v8f acc1 = {0,0,0,0,0,0,0,0};
  const int half = lane >> 4;     // 0: lanes 0-15 (K k..k+1), 1: lanes 16-31 (K k+2..k+3)
  const int lmod = lane & 15;
  #pragma unroll
  for (int k0 = 0; k0 < 16; k0 += 4) {
    const int kk = k0 + 2*half;
    // B operand: rows kk,kk+1 of A, column = lmod (broadcast loads via L2)
    v2f bop;
    bop.x = Ag[ kk      * 16 + lmod];
    bop.y = Ag[(kk + 1) * 16 + lmod];
    // A operands: S rows (tile 0: rows 0-15, tile 1: rows 16-31), K = kk,kk+1
    const v2f a0 = *reinterpret_cast<const v2f*>(&Sw[(lmod     ) * 16 + kk]);
    const v2f a1 = *reinterpret_cast<const v2f*>(&Sw[(lmod + 16) * 16 + kk]);
    acc0 = __builtin_amdgcn_wmma_f32_16x16x4_f32(false, a0, false, bop, (short)0, acc0, false, false);
    acc1 = __builtin_amdgcn_wmma_f32_16x16x4_f32(false, a1, false, bop, (short)0, acc1, false, false);
  }

  // ---- scatter P (C/D layout: VGPR v -> row v + 8*half, col lmod) to LDS ----
  float* Pw = &Pbuf[wv][0];
  #pragma unroll
  for (int v = 0; v < 8; v++) {
    Pw[(     v + 8*half) * 16 + lmod] = acc0[v];
    Pw[(16 + v + 8*half) * 16 + lmod] = acc1[v];
  }
  asm volatile("" ::: "memory");

  // ---- out[b] = dot(P[row=lane], s) with s still in registers ----
  const float4* prow = reinterpret_cast<const float4*>(&Pw[lane * 16]);
  float qa = 0.f, qb = 0.f, qc = 0.f, qd = 0.f;
  #pragma unroll
  for (int i = 0; i < 4; i++) {
    const float4 pv = prow[i];
    qa += pv.x * sv[4*i + 0];
    qb += pv.y * sv[4*i + 1];
    qc += pv.z * sv[4*i + 2];
    qd += pv.w * sv[4*i + 3];
  }
  out[b] = (qa + qb) + (qc + qd);
}

// ---------------------------------------------------------------------------
extern "C" void kernel_launch(void* const* d_in, const int* in_sizes, int n_in,
                              void* d_out, int out_size, void* d_ws, size_t ws_size,
                              hipStream_t stream) {
  const float* x = (const float*)d_in[0];   // [B,4]
  const float* w = (const float*)d_in[1];   // [1,4,3]
  float* Ag  = (float*)d_ws;                // 16x16 f32
  float* out = (float*)d_out;               // [B]
  const int B = in_sizes[0] / 4;            // 2,097,152 (multiple of 256)

  hipLaunchKernelGGL(qc_setup_kernel, dim3(1), dim3(32), 0, stream, w, Ag);
  const int threads = 256;
  const int blocks  = B / threads;
  hipLaunchKernelGGL(qc_main_kernel, dim3(blocks), dim3(threads), 0, stream, x, Ag, out);
}